// HGATLayer_39307540693803
// MI455X (gfx1250) — compile-verified
//
#include <hip/hip_runtime.h>

typedef __attribute__((ext_vector_type(2))) float v2f;
typedef __attribute__((ext_vector_type(4))) float v4f;
typedef __attribute__((ext_vector_type(8))) float v8f;

#define DIN 128
#define DOUT 128

// ---------------------------------------------------------------- zero fill
__global__ void hg_zero(float* __restrict__ p, long n) {
  long i = (long)blockIdx.x * blockDim.x + threadIdx.x;
  const long stride = (long)gridDim.x * blockDim.x;
  for (; i < n; i += stride) p[i] = 0.0f;
}

// ------------------------------------------------------------ degree counts
// D[node] += 1, B[edge] += 1 via global_atomic_add_f32
__global__ void hg_degrees(const long long* __restrict__ nidx,
                           const long long* __restrict__ eidx,
                           float* __restrict__ D, float* __restrict__ B,
                           int nnz) {
  int i = blockIdx.x * blockDim.x + threadIdx.x;
  if (i < nnz) {
    atomicAdd(D + (int)nidx[i], 1.0f);
    atomicAdd(B + (int)eidx[i], 1.0f);
  }
}

// ------------------------------------------------- deg -> 1/deg in place
__global__ void hg_inv(float* __restrict__ deg, int n) {
  int i = blockIdx.x * blockDim.x + threadIdx.x;
  if (i < n) {
    const float d = deg[i];
    deg[i] = d > 0.0f ? 1.0f / d : 0.0f;
  }
}

// ---------------------------------------------------- h = x @ W (fp32 WMMA)
// Block = 256 threads = 8 waves. Block owns 16 rows of x; wave w owns the
// 16x16 output tile at columns [16w, 16w+16). W (64 KB) staged in LDS.
// A-frag (16x4 f32, 2 VGPRs/lane): lanes 0-15 -> M=lane, K=k0+{0,1};
//                                  lanes 16-31 -> M=lane-16, K=k0+{2,3}.
// B-frag (4x16 f32): same half-wave K split, N = lane%16.
// C/D (16x16 f32, 8 VGPRs): vgpr v -> M = v + 8*(lane>=16), N = lane%16.
__global__ __launch_bounds__(256) void hg_gemm_wmma(
    const float* __restrict__ x, const float* __restrict__ w,
    float* __restrict__ h, int nrows) {
  __shared__ float lw[DIN * DOUT];  // 64 KB of the WGP's 320 KB LDS
  {
    const v4f* src = (const v4f*)w;
    v4f* dst = (v4f*)lw;
    const int t = threadIdx.x;
#pragma unroll
    for (int i = 0; i < (DIN * DOUT / 4) / 256; ++i)
      dst[t + i * 256] = src[t + i * 256];
  }
  __syncthreads();

  const int lane  = threadIdx.x & 31;
  const int wv    = threadIdx.x >> 5;
  const int lhalf = lane >> 4;   // 0: K+{0,1}, 1: K+{2,3}
  const int l16   = lane & 15;
  const int m0    = blockIdx.x * 16;
  const int n0    = wv * 16;

  int arowi = m0 + l16;
  if (arowi >= nrows) arowi = nrows - 1;  // clamp loads; EXEC stays all-ones
  const float* __restrict__ arow = x + (size_t)arowi * DIN;

  v8f c = {};
#pragma unroll
  for (int k0 = 0; k0 < DIN; k0 += 4) {
    const int ka = k0 + lhalf * 2;
    v2f a, b;
    a.x = arow[ka + 0];
    a.y = arow[ka + 1];
    b.x = lw[(ka + 0) * DOUT + n0 + l16];
    b.y = lw[(ka + 1) * DOUT + n0 + l16];
    // (neg_a, A, neg_b, B, c_mod, C, reuse_a, reuse_b)
    c = __builtin_amdgcn_wmma_f32_16x16x4_f32(false, a, false, b, (short)0, c,
                                              false, false);
  }

  float* __restrict__ hp = h + (size_t)m0 * DOUT + n0 + l16;
#pragma unroll
  for (int v = 0; v < 8; ++v) {
    const int m = v + lhalf * 8;
    if (m0 + m < nrows) hp[(size_t)m * DOUT] = c[v];
  }
}

// ---------------------------------------------- segment-sum scatter (rows)
// One wave per incidence entry; lane handles the float4 at column 4*lane.
// 1x global_load_b128 + 4x global_atomic_add_f32 per lane; rows live in L2.
// Optional per-gather-row scale (inverse degree) fused at gather time.
__global__ __launch_bounds__(256) void hg_scatter(
    const float* __restrict__ src,              // [*, 128] gathered rows
    const long long* __restrict__ gather_idx,   // row of src per entry
    const long long* __restrict__ scatter_idx,  // row of dst per entry
    const float* __restrict__ gscale,           // 1/deg per src row, or null
    float* __restrict__ dst, int nnz) {
  const int e = blockIdx.x * 8 + (threadIdx.x >> 5);
  const int lane = threadIdx.x & 31;
  if (e >= nnz) return;
  const long long gi = gather_idx[e];
  const size_t srow = (size_t)gi * DOUT;
  const size_t drow = (size_t)scatter_idx[e] * DOUT;
  const float s = gscale ? gscale[gi] : 1.0f;
  v4f val = *(const v4f*)(src + srow + lane * 4);
  val.x *= s; val.y *= s; val.z *= s; val.w *= s;
  float* dp = dst + drow + lane * 4;
  atomicAdd(dp + 0, val.x);
  atomicAdd(dp + 1, val.y);
  atomicAdd(dp + 2, val.z);
  atomicAdd(dp + 3, val.w);
}

// ------------------------------------- out = out * Dinv[row] + bias[col]
__global__ void hg_final(float* __restrict__ out,
                         const float* __restrict__ dinv,
                         const float* __restrict__ bias, long nelem) {
  long i = (long)blockIdx.x * blockDim.x + threadIdx.x;
  if (i < nelem) {
    out[i] = out[i] * dinv[i >> 7] + bias[i & (DOUT - 1)];  // DOUT == 128
  }
}

// ---------------------------------------------------------------- launcher
extern "C" void kernel_launch(void* const* d_in, const int* in_sizes, int n_in,
                              void* d_out, int out_size, void* d_ws,
                              size_t ws_size, hipStream_t stream) {
  const float* x        = (const float*)d_in[0];      // [N, 128]
  const float* w        = (const float*)d_in[1];      // [128, 128]
  const float* bias     = (const float*)d_in[2];      // [128]
  const long long* hei  = (const long long*)d_in[3];  // [2, NNZ] int64
  float* out            = (float*)d_out;              // [N, 128]

  const int N   = in_sizes[0] / DIN;
  const int NNZ = in_sizes[3] / 2;
  const int E   = N;  // edge ids drawn from [0, N) in this problem

  const long long* nidx = hei;        // hyperedge_index[0]
  const long long* eidx = hei + NNZ;  // hyperedge_index[1]

  // workspace layout: h | e_feat | D | B   (~103 MB total)
  float* h     = (float*)d_ws;
  float* efeat = h + (size_t)N * DOUT;
  float* D     = efeat + (size_t)E * DOUT;
  float* B     = D + N;

  const long nND = (long)N * DOUT;
  const long nED = (long)E * DOUT;

  hg_zero<<<2048, 256, 0, stream>>>(efeat, nED);
  hg_zero<<<256, 256, 0, stream>>>(D, (long)N);
  hg_zero<<<256, 256, 0, stream>>>(B, (long)E);
  hg_zero<<<2048, 256, 0, stream>>>(out, nND);

  hg_degrees<<<(NNZ + 255) / 256, 256, 0, stream>>>(nidx, eidx, D, B, NNZ);
  // convert degrees to inverse degrees in place (tiny passes)
  hg_inv<<<(N + 255) / 256, 256, 0, stream>>>(D, N);
  hg_inv<<<(E + 255) / 256, 256, 0, stream>>>(B, E);

  hg_gemm_wmma<<<(N + 15) / 16, 256, 0, stream>>>(x, w, h, N);

  // nodes -> hyperedges (unscaled gather of h)
  hg_scatter<<<(NNZ + 7) / 8, 256, 0, stream>>>(h, nidx, eidx, nullptr, efeat,
                                                NNZ);
  // hyperedges -> nodes, Binv fused at gather time
  hg_scatter<<<(NNZ + 7) / 8, 256, 0, stream>>>(efeat, eidx, nidx, B, out,
                                                NNZ);
  // out = out * Dinv + bias
  hg_final<<<(unsigned)((nND + 255) / 256), 256, 0, stream>>>(out, D, bias,
                                                              nND);
}